// GCN_83038897701147
// MI455X (gfx1250) — compile-verified
//
#include <hip/hip_runtime.h>

typedef __attribute__((ext_vector_type(2))) float v2f;
typedef __attribute__((ext_vector_type(8))) float v8f;

#define DFEAT 64
#define THREADS 256
#define WSTRIDE 160   // words per K-pair row in LDS; 160 % 64 == 32 -> adjacent
                      // K-pair rows land on disjoint bank halves (conflict-free b64)

// -------------------- utility kernels --------------------

__global__ __launch_bounds__(THREADS) void zero_f32(float* __restrict__ p, int n) {
    int i = blockIdx.x * THREADS + threadIdx.x;
    if (i < n) p[i] = 0.0f;
}

__global__ __launch_bounds__(THREADS) void degree_kernel(const int* __restrict__ src,
                                                         const int* __restrict__ dst,
                                                         float* __restrict__ deg_out,
                                                         float* __restrict__ deg_in,
                                                         int E) {
    int e = blockIdx.x * THREADS + threadIdx.x;
    if (e < E) {
        atomicAdd(deg_out + src[e], 1.0f);
        atomicAdd(deg_in  + dst[e], 1.0f);
    }
}

__global__ __launch_bounds__(THREADS) void rsqrt_inplace(float* __restrict__ p, int n) {
    int i = blockIdx.x * THREADS + threadIdx.x;
    if (i < n) p[i] = rsqrtf(fmaxf(p[i], 1.0f));
}

// -------------------- fp32 WMMA GEMM: H = (X * norm_src[:,None]) @ W --------------------
// Block: 256 threads = 8 waves. Each wave computes a 16-row strip of H across all
// 64 output columns (4 accumulators of 16x16). W staged in LDS K-pair-interleaved:
//   Wl[(k>>1)*WSTRIDE + n*2 + (k&1)] = W[k][n]
// so each B fragment {W[k][n], W[k+1][n]} is ONE aligned ds_load_b64 straight into
// the even-aligned VGPR pair V_WMMA_F32_16X16X4_F32 consumes (no repack movs).

__global__ __launch_bounds__(THREADS) void gemm_norm_wmma(const float* __restrict__ X,
                                                          const float* __restrict__ norm_src,
                                                          const float* __restrict__ W,
                                                          float* __restrict__ H,
                                                          int n_rows) {
    __shared__ float Wl[32 * WSTRIDE];   // 20 KB

    // stage W interleaved (1024 float4 loads across 256 threads, 4 passes)
    #pragma unroll
    for (int p = 0; p < 4; ++p) {
        int i = threadIdx.x + p * THREADS;          // i in [0,1024)
        float4 w4 = ((const float4*)W)[i];
        int k  = i >> 4;                            // row of W (K index)
        int n0 = (i & 15) * 4;                      // starting column
        float* dp = Wl + (k >> 1) * WSTRIDE + (k & 1);
        dp[(n0 + 0) * 2] = w4.x;
        dp[(n0 + 1) * 2] = w4.y;
        dp[(n0 + 2) * 2] = w4.z;
        dp[(n0 + 3) * 2] = w4.w;
    }
    __syncthreads();

    const int wave = threadIdx.x >> 5;
    const int lane = threadIdx.x & 31;
    const int m0 = blockIdx.x * 128 + wave * 16;
    if (m0 >= n_rows) return;   // wave-uniform: EXEC stays all-1s for WMMA

    const int mrow  = m0 + (lane & 15);
    const int kband = (lane >> 4) << 1;       // 0 or 2 (A/B K-slot per half-wave)
    const int ncol  = lane & 15;

    const float nv = norm_src[mrow];
    const float* Xr = X + (size_t)mrow * DFEAT;

    v8f acc0 = {}, acc1 = {}, acc2 = {}, acc3 = {};

    #pragma unroll
    for (int k0 = 0; k0 < DFEAT; k0 += 4) {
        const int k = k0 + kband;                 // even
        // A fragment: rows m0..m0+15, K slots k..k+1 (fused norm scaling)
        float2 xa = *(const float2*)(Xr + k);
        v2f a;
        a.x = xa.x * nv;
        a.y = xa.y * nv;

        // B fragments: one ds_load_b64 each, bank-conflict-free
        const v2f* Bp = (const v2f*)(Wl + (k >> 1) * WSTRIDE);
        acc0 = __builtin_amdgcn_wmma_f32_16x16x4_f32(false, a, false, Bp[ncol],      (short)0, acc0, false, false);
        acc1 = __builtin_amdgcn_wmma_f32_16x16x4_f32(false, a, false, Bp[16 + ncol], (short)0, acc1, false, false);
        acc2 = __builtin_amdgcn_wmma_f32_16x16x4_f32(false, a, false, Bp[32 + ncol], (short)0, acc2, false, false);
        acc3 = __builtin_amdgcn_wmma_f32_16x16x4_f32(false, a, false, Bp[48 + ncol], (short)0, acc3, false, false);
    }

    // D layout: VGPR i, lanes 0-15 -> M=i, lanes 16-31 -> M=i+8
    const int rsel = (lane >> 4) * 8;
    #pragma unroll
    for (int i = 0; i < 8; ++i) {
        float* Hr = H + (size_t)(m0 + rsel + i) * DFEAT + ncol;
        Hr[0]  = acc0[i];
        Hr[16] = acc1[i];
        Hr[32] = acc2[i];
        Hr[48] = acc3[i];
    }
}

// -------------------- edge scatter: AGG[dst] += H[src] --------------------
// 16 threads per edge, each handles 4 consecutive features (float4 gather,
// 4x global_atomic_add_f32 scatter). Coalesced on the gather side; H/AGG are
// 25.6 MB each and stay resident in the 192 MB L2.

__global__ __launch_bounds__(THREADS) void scatter_add(const float* __restrict__ H,
                                                       const int* __restrict__ src,
                                                       const int* __restrict__ dst,
                                                       float* __restrict__ AGG,
                                                       int E) {
    long long idx = (long long)blockIdx.x * THREADS + threadIdx.x;
    int e = (int)(idx >> 4);
    if (e >= E) return;
    int t = (int)(idx & 15);
    int s = src[e];
    int d = dst[e];
    float4 v = ((const float4*)(H + (size_t)s * DFEAT))[t];
    float* o = AGG + (size_t)d * DFEAT + t * 4;
    atomicAdd(o + 0, v.x);
    atomicAdd(o + 1, v.y);
    atomicAdd(o + 2, v.z);
    atomicAdd(o + 3, v.w);
}

// -------------------- finalize: out = relu(AGG*norm_dst + b) + res --------------------

__global__ __launch_bounds__(THREADS) void finalize_kernel(const float* __restrict__ AGG,
                                                           const float* __restrict__ norm_dst,
                                                           const float* __restrict__ bias,
                                                           const float* __restrict__ res,
                                                           float* __restrict__ out,
                                                           int total) {
    int idx = blockIdx.x * THREADS + threadIdx.x;
    if (idx >= total) return;
    int n = idx >> 6;
    int c = idx & 63;
    float v = AGG[idx] * norm_dst[n] + bias[c];
    out[idx] = fmaxf(v, 0.0f) + res[idx];
}

// -------------------- host orchestration --------------------

static void run_layer(const float* xin, const float* Wm, const float* bias,
                      const int* src, const int* dst,
                      const float* nsrc, const float* ndst,
                      float* H, float* AGG, float* xout,
                      int N, int E, hipStream_t stream) {
    int totalNF = N * DFEAT;
    gemm_norm_wmma<<<(N + 127) / 128, THREADS, 0, stream>>>(xin, nsrc, Wm, H, N);
    zero_f32<<<(totalNF + THREADS - 1) / THREADS, THREADS, 0, stream>>>(AGG, totalNF);
    long long sthreads = (long long)E * 16;
    scatter_add<<<(unsigned)((sthreads + THREADS - 1) / THREADS), THREADS, 0, stream>>>(
        H, src, dst, AGG, E);
    finalize_kernel<<<(totalNF + THREADS - 1) / THREADS, THREADS, 0, stream>>>(
        AGG, ndst, bias, xin, xout, totalNF);
}

extern "C" void kernel_launch(void* const* d_in, const int* in_sizes, int n_in,
                              void* d_out, int out_size, void* d_ws, size_t ws_size,
                              hipStream_t stream) {
    const float* feat = (const float*)d_in[0];
    const int*   src  = (const int*)d_in[1];
    const int*   dst  = (const int*)d_in[2];
    // d_in[3] = etype (unused by reference math)
    const float* w1 = (const float*)d_in[4];
    const float* b1 = (const float*)d_in[5];
    const float* w2 = (const float*)d_in[6];
    const float* b2 = (const float*)d_in[7];
    const float* w3 = (const float*)d_in[8];
    const float* b3 = (const float*)d_in[9];

    const int N = in_sizes[0] / DFEAT;
    const int E = in_sizes[1];

    // workspace layout (floats): norm_src[N] | norm_dst[N] | H[N*64] | AGG[N*64] | X[N*64]
    float* ws   = (float*)d_ws;
    float* nsrc = ws;
    float* ndst = nsrc + N;
    float* H    = ndst + N;
    float* AGG  = H + (size_t)N * DFEAT;
    float* X    = AGG + (size_t)N * DFEAT;
    float* out  = (float*)d_out;

    // degrees -> norms (deg accumulated in the norm buffers, then rsqrt in place)
    zero_f32<<<(2 * N + THREADS - 1) / THREADS, THREADS, 0, stream>>>(nsrc, 2 * N);
    degree_kernel<<<(E + THREADS - 1) / THREADS, THREADS, 0, stream>>>(src, dst, nsrc, ndst, E);
    rsqrt_inplace<<<(2 * N + THREADS - 1) / THREADS, THREADS, 0, stream>>>(nsrc, 2 * N);

    // three GraphConv layers with relu + residual
    run_layer(feat, w1, b1, src, dst, nsrc, ndst, H, AGG, X,   N, E, stream);
    run_layer(X,    w2, b2, src, dst, nsrc, ndst, H, AGG, X,   N, E, stream);
    run_layer(X,    w3, b3, src, dst, nsrc, ndst, H, AGG, out, N, E, stream);
}